// Generator3DLUT_identity_13812614824357
// MI455X (gfx1250) — compile-verified
//
#include <hip/hip_runtime.h>
#include <math.h>

#define LDS_AS __attribute__((address_space(3)))

typedef float v4f __attribute__((ext_vector_type(4)));

namespace {
constexpr int       DIM = 33;
constexpr int       D2  = DIM * DIM;        // 1089
constexpr int       D3  = DIM * DIM * DIM;  // 35937
constexpr long long HW  = 1024LL * 1024LL;  // pixels per plane
constexpr long long GPP = HW / 4;           // float4 groups per plane (2^18)
}

// ---------------------------------------------------------------------------
// Pre-pass: repack planar LUT (3 x 33^3 floats) into interleaved float4
// entries so every trilinear tap is a single global_load_b128 (L2-resident).
// ---------------------------------------------------------------------------
__global__ void repack_lut_kernel(const float* __restrict__ lut,
                                  float4* __restrict__ lut4, int d3)
{
    int i = blockIdx.x * blockDim.x + threadIdx.x;
    if (i < d3) {
        float4 v;
        v.x = lut[i];
        v.y = lut[i + d3];
        v.z = lut[i + 2 * d3];
        v.w = 0.0f;
        lut4[i] = v;
    }
}

// ---------------------------------------------------------------------------
// Trilinear sample of one pixel (exactly mirrors the reference math).
// ---------------------------------------------------------------------------
template <bool PACKED>
__device__ __forceinline__ void lut_sample1(float r, float g, float b,
                                            const float4* __restrict__ lut4,
                                            const float* __restrict__ lut,
                                            float& oR, float& oG, float& oB)
{
    const float invbin = (float)(DIM - 1) / 1.0001f;  // 1/binsize
    float rs = r * invbin, gs = g * invbin, bs = b * invbin;
    int ri = (int)floorf(rs), gi = (int)floorf(gs), bi = (int)floorf(bs);
    ri = ri < 0 ? 0 : (ri > DIM - 2 ? DIM - 2 : ri);
    gi = gi < 0 ? 0 : (gi > DIM - 2 ? DIM - 2 : gi);
    bi = bi < 0 ? 0 : (bi > DIM - 2 ? DIM - 2 : bi);
    float rd = rs - (float)ri, gd = gs - (float)gi, bd = bs - (float)bi;
    float wr[2] = {1.0f - rd, rd};
    float wg[2] = {1.0f - gd, gd};
    float wb[2] = {1.0f - bd, bd};
    int base = ri + gi * DIM + bi * D2;
    float aR = 0.f, aG = 0.f, aB = 0.f;
#pragma unroll
    for (int db = 0; db < 2; ++db)
#pragma unroll
        for (int dg = 0; dg < 2; ++dg)
#pragma unroll
            for (int dr = 0; dr < 2; ++dr) {
                float w  = wr[dr] * wg[dg] * wb[db];
                int  idx = base + dr + dg * DIM + db * D2;
                float vr, vg, vb;
                if (PACKED) {
                    float4 v = lut4[idx];   // one b128 gather per tap
                    vr = v.x; vg = v.y; vb = v.z;
                } else {
                    vr = lut[idx];
                    vg = lut[idx + D3];
                    vb = lut[idx + 2 * D3];
                }
                aR += w * vr; aG += w * vg; aB += w * vb;
            }
    oR = aR; oG = aG; oB = aB;
}

template <bool PACKED>
__device__ __forceinline__ void lut_sample4(const float4& r4, const float4& g4,
                                            const float4& b4,
                                            const float4* __restrict__ lut4,
                                            const float* __restrict__ lut,
                                            float4& oR, float4& oG, float4& oB)
{
    lut_sample1<PACKED>(r4.x, g4.x, b4.x, lut4, lut, oR.x, oG.x, oB.x);
    lut_sample1<PACKED>(r4.y, g4.y, b4.y, lut4, lut, oR.y, oG.y, oB.y);
    lut_sample1<PACKED>(r4.z, g4.z, b4.z, lut4, lut, oR.z, oG.z, oB.z);
    lut_sample1<PACKED>(r4.w, g4.w, b4.w, lut4, lut, oR.w, oG.w, oB.w);
}

// Non-temporal float4 store: output is write-once streaming data; keep it
// from displacing the L2-resident LUT.
__device__ __forceinline__ void store_nt4(float* q, const float4& v)
{
    v4f tmp;
    tmp.x = v.x; tmp.y = v.y; tmp.z = v.z; tmp.w = v.w;
    __builtin_nontemporal_store(tmp, (v4f*)q);
}

#if defined(__gfx1250__)
// Async global -> LDS B128 copy (CDNA5 ASYNCcnt path). Raw gfx1250
// instruction via inline asm: bypasses the clang builtin's LangAS-qualified
// prototype (which cannot be spelled in HIP source).
//   vdst  = VGPR holding the wave-relative LDS byte address
//   vaddr = VGPR pair holding the 64-bit global address (GV mode)
// th:TH_LOAD_NT: input pixels are read exactly once -> non-temporal.
__device__ __forceinline__ void async_copy_b128(void* lds, const void* gsrc)
{
    unsigned           ldsoff = (unsigned)(unsigned long long)(LDS_AS char*)lds;
    unsigned long long ga     = (unsigned long long)gsrc;
    asm volatile("global_load_async_to_lds_b128 %0, %1, off th:TH_LOAD_NT"
                 :: "v"(ldsoff), "v"(ga)
                 : "memory");
}
#endif

// ---------------------------------------------------------------------------
// Main kernel: each thread owns one float4 group (4 pixels) per iteration.
// gfx1250 path: double-buffered async global->LDS prefetch of the next tile
// while gathering/interpolating the current one. No cross-wave LDS sharing,
// so synchronization is s_wait_asynccnt only.
// ---------------------------------------------------------------------------
template <bool PACKED>
__global__ __launch_bounds__(256) void apply_lut_kernel(
    const float* __restrict__ x,
    const float* __restrict__ lut,
    const float4* __restrict__ lut4,
    float* __restrict__ out,
    long long ngroups, int iters)
{
    const int       t     = threadIdx.x;
    const long long gbase = (long long)blockIdx.x * iters * 256 + t;

#if defined(__gfx1250__)
    __shared__ float4 lbuf[2][3][256];  // 24 KB of 320 KB LDS

    auto issue = [&](int buf, long long g) {
        long long gc  = g < ngroups ? g : (ngroups - 1);
        long long n   = gc / GPP;
        size_t    pos = (size_t)(gc % GPP) * 4;
        const float* p0 = x + (size_t)(n * 3) * HW + pos;
        async_copy_b128(&lbuf[buf][0][t], p0);
        async_copy_b128(&lbuf[buf][1][t], p0 + HW);
        async_copy_b128(&lbuf[buf][2][t], p0 + 2 * HW);
    };

    issue(0, gbase);  // prologue

    for (int it = 0; it < iters; ++it) {
        long long g = gbase + 256LL * it;
        if (it + 1 < iters) {
            // buffer being refilled was consumed last iteration; async LDS
            // writes are unordered vs our DS reads, so drain DScnt first.
            asm volatile("s_wait_dscnt 0" ::: "memory");
            issue((it + 1) & 1, g + 256);
            asm volatile("s_wait_asynccnt 3" ::: "memory");  // current tile done
        } else {
            asm volatile("s_wait_asynccnt 0" ::: "memory");
        }
        const int cb = it & 1;
        float4 r4 = lbuf[cb][0][t];
        float4 g4 = lbuf[cb][1][t];
        float4 b4 = lbuf[cb][2][t];
        float4 oR, oG, oB;
        lut_sample4<PACKED>(r4, g4, b4, lut4, lut, oR, oG, oB);
        if (g < ngroups) {
            long long n   = g / GPP;
            size_t    pos = (size_t)(g % GPP) * 4;
            float*    q   = out + (size_t)(n * 3) * HW + pos;
            store_nt4(q, oR);
            store_nt4(q + HW, oG);
            store_nt4(q + 2 * HW, oB);
        }
    }
#else
    for (int it = 0; it < iters; ++it) {
        long long g = gbase + 256LL * it;
        if (g >= ngroups) continue;
        long long n   = g / GPP;
        size_t    pos = (size_t)(g % GPP) * 4;
        const float* p0 = x + (size_t)(n * 3) * HW + pos;
        float4 r4 = *(const float4*)(p0);
        float4 g4 = *(const float4*)(p0 + HW);
        float4 b4 = *(const float4*)(p0 + 2 * HW);
        float4 oR, oG, oB;
        lut_sample4<PACKED>(r4, g4, b4, lut4, lut, oR, oG, oB);
        float* q = out + (size_t)(n * 3) * HW + pos;
        store_nt4(q, oR);
        store_nt4(q + HW, oG);
        store_nt4(q + 2 * HW, oB);
    }
#endif
}

// ---------------------------------------------------------------------------
// Host launcher
// ---------------------------------------------------------------------------
extern "C" void kernel_launch(void* const* d_in, const int* in_sizes, int n_in,
                              void* d_out, int out_size, void* d_ws, size_t ws_size,
                              hipStream_t stream)
{
    const float* LUT = (const float*)d_in[0];  // (3, 33, 33, 33)
    const float* x   = (const float*)d_in[1];  // (N, 3, 1024, 1024)
    float*       out = (float*)d_out;          // (N, 3, 1024, 1024)

    const int       d3      = in_sizes[0] / 3;                 // 35937
    const long long N       = (long long)in_sizes[1] / (3 * HW);
    const long long ngroups = N * GPP;

    const bool packed = ws_size >= (size_t)d3 * sizeof(float4);
    float4*    lut4   = (float4*)d_ws;

    if (packed) {
        repack_lut_kernel<<<(d3 + 255) / 256, 256, 0, stream>>>(LUT, lut4, d3);
    }

    const int       iters  = 4;
    const long long blocks = (ngroups + 256LL * iters - 1) / (256LL * iters);

    if (packed) {
        apply_lut_kernel<true><<<(int)blocks, 256, 0, stream>>>(
            x, LUT, lut4, out, ngroups, iters);
    } else {
        apply_lut_kernel<false><<<(int)blocks, 256, 0, stream>>>(
            x, LUT, lut4, out, ngroups, iters);
    }
}